// RauchTungStriebelSmoother_50946902065703
// MI455X (gfx1250) — compile-verified
//
#include <hip/hip_runtime.h>

// RTS smoother for B=64, T=512, D=64 on gfx1250 (MI455X).
// Phase 1: parallel gains G_t = Pf Ft inv(F Pf Ft + Q) via FP32 WMMA + Gauss-Jordan.
// Phase 2: sequential affine scan per batch using only WMMA matmuls (no inverse),
//          with global_prefetch of the next step's tiles and NT cache policy on
//          the streaming (>L2-sized) arrays.

constexpr int B = 64;
constexpr int T = 512;
constexpr int D = 64;               // state dim (64x64 matrices)
constexpr int MAT = D * D;          // 4096 floats per matrix
constexpr int NT = 512;             // threads per block = 16 waves (wave32)
constexpr size_t XS_TOTAL = (size_t)B * T * D;

typedef float v2f __attribute__((ext_vector_type(2)));
typedef float v8f __attribute__((ext_vector_type(8)));

// 64x64 FP32 matmul on LDS operands using V_WMMA_F32_16X16X4_F32.
// Cout = A @ B (+ Cin), optionally B transposed. 16 waves, one 16x16 tile each.
__device__ __forceinline__ void mm64(const float* A, const float* Bm,
                                     const float* Cin, float* Cout, bool transB) {
  const int tid  = threadIdx.x;
  const int wave = tid >> 5;
  const int lane = tid & 31;
  const int tm   = (wave >> 2) << 4;   // tile row base
  const int tn   = (wave & 3) << 4;    // tile col base
  const int half = lane >> 4;          // 0: lanes 0-15, 1: lanes 16-31
  const int l16  = lane & 15;
  const int kb   = half << 1;          // K offset within 4-chunk: 0 or 2

  const int crow = tm + (half << 3);   // C/D: VGPR r -> row crow+r
  const int ccol = tn + l16;

  v8f c;
  if (Cin) {
#pragma unroll
    for (int r = 0; r < 8; ++r) c[r] = Cin[(crow + r) * D + ccol];
  } else {
#pragma unroll
    for (int r = 0; r < 8; ++r) c[r] = 0.0f;
  }

#pragma unroll
  for (int kk = 0; kk < D; kk += 4) {
    const int k = kk + kb;
    v2f a, b;
    // A fragment: lane = row, VGPR0/1 = K = kb, kb+1 (ISA 16x4 f32 layout)
    a.x = A[(tm + l16) * D + k];
    a.y = A[(tm + l16) * D + k + 1];
    // B fragment: lane = col, VGPR0/1 = K = kb, kb+1
    if (!transB) {
      b.x = Bm[(k)     * D + ccol];
      b.y = Bm[(k + 1) * D + ccol];
    } else {                            // use B^T: B^T[k][n] = B[n][k]
      b.x = Bm[ccol * D + k];
      b.y = Bm[ccol * D + k + 1];
    }
    c = __builtin_amdgcn_wmma_f32_16x16x4_f32(false, a, false, b,
                                              (short)0, c, false, false);
  }

#pragma unroll
  for (int r = 0; r < 8; ++r) Cout[(crow + r) * D + ccol] = c[r];
}

// In-place Gauss-Jordan inverse of SPD 64x64 matrix in LDS (no pivoting needed).
__device__ __forceinline__ void invert64(float* M, float* srow, float* scol) {
  const int tid = threadIdx.x;
  for (int k = 0; k < D; ++k) {
    __syncthreads();
    if (tid < D) { srow[tid] = M[k * D + tid]; scol[tid] = M[tid * D + k]; }
    __syncthreads();
    const float invp = 1.0f / srow[k];
#pragma unroll
    for (int r = 0; r < 8; ++r) {
      const int idx = tid + (r << 9);
      const int i = idx >> 6, j = idx & 63;
      float v;
      if (i == k)      v = (j == k) ? invp : srow[j] * invp;
      else if (j == k) v = -scol[i] * invp;
      else             v = M[idx] - scol[i] * invp * srow[j];
      M[idx] = v;
    }
  }
  __syncthreads();
}

// ---------------- Phase 1: gains (parallel over b,t) ----------------
__global__ __launch_bounds__(NT)
void rts_gain_kernel(const float* __restrict__ Pf, const float* __restrict__ F,
                     const float* __restrict__ Q, float* __restrict__ out_Ps) {
  __shared__ float sF[MAT], sQ[MAT], sPi[MAT], sA[MAT], sB[MAT];
  __shared__ float srow[D], scol[D];

  const int tid = threadIdx.x;
  const size_t bt = blockIdx.x;              // 0 .. B*(T-1)-1
  const int b = (int)(bt / (T - 1));
  const int t = (int)(bt % (T - 1));         // gains needed for t in [0, T-2]
  const float* Pi = Pf + ((size_t)b * T + t) * MAT;

#pragma unroll
  for (int r = 0; r < 8; ++r) {
    const int idx = tid + (r << 9);
    sF[idx]  = F[idx];
    sQ[idx]  = Q[idx];
    sPi[idx] = __builtin_nontemporal_load(&Pi[idx]);   // 512MB stream >> 192MB L2
  }
  __syncthreads();

  mm64(sF,  sPi, nullptr, sA, false);  __syncthreads();   // T1 = F @ Pi
  mm64(sA,  sF,  sQ,      sB, true);   __syncthreads();   // Pp = T1 @ F^T + Q
  mm64(sPi, sF,  nullptr, sA, true);   __syncthreads();   // T2 = Pi @ F^T
  invert64(sB, srow, scol);                               // sB = inv(Pp)
  mm64(sA,  sB,  nullptr, sPi, false); __syncthreads();   // G = T2 @ inv(Pp)

  // Stage G into the out_Ps slot for time t (consumed then overwritten in phase 2).
  float* Gout = out_Ps + ((size_t)b * T + t) * MAT;
#pragma unroll
  for (int r = 0; r < 8; ++r) {
    const int idx = tid + (r << 9);
    __builtin_nontemporal_store(sPi[idx], &Gout[idx]);
  }
}

// ---------------- Phase 2: sequential backward scan per batch ----------------
__global__ __launch_bounds__(NT)
void rts_smooth_kernel(const float* __restrict__ xf, const float* __restrict__ Pf,
                       const float* __restrict__ F, const float* __restrict__ Q,
                       float* __restrict__ out_xs, float* __restrict__ out_Ps) {
  __shared__ float sF[MAT], sQ[MAT], sPi[MAT], sG[MAT], sPs[MAT], sA[MAT], sB[MAT];
  __shared__ float sx[D], sxs[D], sv[D];

  const int tid = threadIdx.x;
  const int b = blockIdx.x;

#pragma unroll
  for (int r = 0; r < 8; ++r) {
    const int idx = tid + (r << 9);
    sF[idx] = F[idx];
    sQ[idx] = Q[idx];
  }

  // Init at t = T-1: smoothed == filtered.
  {
    const size_t bt = (size_t)b * T + (T - 1);
#pragma unroll
    for (int r = 0; r < 8; ++r) {
      const int idx = tid + (r << 9);
      const float v = __builtin_nontemporal_load(&Pf[bt * MAT + idx]);
      sPs[idx] = v;
      __builtin_nontemporal_store(v, &out_Ps[bt * MAT + idx]);
    }
    if (tid < D) {
      const float v = xf[bt * D + tid];
      sxs[tid] = v;
      out_xs[bt * D + tid] = v;
    }
  }
  __syncthreads();

  for (int t = T - 2; t >= 0; --t) {
    const size_t bt = (size_t)b * T + t;

    // Load Pi, G (staged by phase 1), x_t.
#pragma unroll
    for (int r = 0; r < 8; ++r) {
      const int idx = tid + (r << 9);
      sPi[idx] = __builtin_nontemporal_load(&Pf[bt * MAT + idx]);
      sG[idx]  = __builtin_nontemporal_load(&out_Ps[bt * MAT + idx]);
    }
    if (tid < D) sx[tid] = xf[bt * D + tid];

    // Prefetch next step's tiles (one global_prefetch_b8 per 128B line) so the
    // HBM latency of the serial chain overlaps the 4 matmuls below.
    if (t > 0 && tid < 128) {
      const size_t btn = bt - 1;
      __builtin_prefetch(((const char*)&Pf[btn * MAT]) + (size_t)tid * 128, 0, 0);
      __builtin_prefetch(((const char*)&out_Ps[btn * MAT]) + (size_t)tid * 128, 0, 0);
    }
    __syncthreads();

    mm64(sF, sPi, nullptr, sA, false); __syncthreads();   // T1 = F @ Pi
    mm64(sA, sF,  sQ,      sB, true);  __syncthreads();   // Pp = T1 @ F^T + Q

    // S = Ps_next - Pp (into sA); v = xs_next - F @ x_t.
#pragma unroll
    for (int r = 0; r < 8; ++r) {
      const int idx = tid + (r << 9);
      sA[idx] = sPs[idx] - sB[idx];
    }
    if (tid < D) {
      float xp = 0.0f;
      for (int j = 0; j < D; ++j) xp += sF[tid * D + j] * sx[j];
      sv[tid] = sxs[tid] - xp;
    }
    __syncthreads();

    mm64(sG, sA, nullptr, sB, false); __syncthreads();    // T3 = G @ S
    mm64(sB, sG, sPi,     sPs, true); __syncthreads();    // Ps = Pi + T3 @ G^T

    // Store Ps (overwrites the G slot); xs = x_t + G @ v.
#pragma unroll
    for (int r = 0; r < 8; ++r) {
      const int idx = tid + (r << 9);
      __builtin_nontemporal_store(sPs[idx], &out_Ps[bt * MAT + idx]);
    }
    if (tid < D) {
      float xn = sx[tid];
      for (int j = 0; j < D; ++j) xn += sG[tid * D + j] * sv[j];
      sxs[tid] = xn;
      out_xs[bt * D + tid] = xn;
    }
    __syncthreads();
  }
}

extern "C" void kernel_launch(void* const* d_in, const int* in_sizes, int n_in,
                              void* d_out, int out_size, void* d_ws, size_t ws_size,
                              hipStream_t stream) {
  const float* xf = (const float*)d_in[0];   // (B,T,D,1)
  const float* Pf = (const float*)d_in[1];   // (B,T,D,D)
  const float* F  = (const float*)d_in[2];   // (D,D)
  const float* Q  = (const float*)d_in[3];   // (D,D)

  float* out_xs = (float*)d_out;             // (B,T,D,1) flat
  float* out_Ps = out_xs + XS_TOTAL;         // (B,T,D,D) flat

  rts_gain_kernel<<<B * (T - 1), NT, 0, stream>>>(Pf, F, Q, out_Ps);
  rts_smooth_kernel<<<B, NT, 0, stream>>>(xf, Pf, F, Q, out_xs, out_Ps);
}